// AutoCorrelationLayer_56770877718643
// MI455X (gfx1250) — compile-verified
//
#include <hip/hip_runtime.h>
#include <math.h>

// Problem constants (fixed by the reference)
#define B_    8
#define L_    1024
#define DM    256
#define H_    8
#define DK_   32
#define KTOP  6
#define NOUT  (L_ * DM)   // 262144 output columns per batch

// Workspace layout (float offsets)
#define OFF_WQ    0        // 256*8 colsums of Wq per head
#define OFF_WK    2048
#define OFF_BQ    4096     // 8
#define OFF_BK    4104     // 8
#define OFF_CORR  4112     // B*L = 8192
#define OFF_W     12304    // 8*6 softmax weights
#define OFF_AGG   12352    // 16*256 (rows 8..15 zero padding for WMMA)
#define OFF_IDX   16448    // 8*6 ints (reinterpret)

typedef __attribute__((ext_vector_type(2))) float v2f;
typedef __attribute__((ext_vector_type(8))) float v8f;

// ---------------------------------------------------------------------------
// 1) Per-head column sums of Wq/Wk (+ bias sums). 4112 work items, trivial.
// ---------------------------------------------------------------------------
__global__ void k_colsum(const float* __restrict__ Wq, const float* __restrict__ bq,
                         const float* __restrict__ Wk, const float* __restrict__ bk,
                         float* __restrict__ ws) {
    int tid = blockIdx.x * 256 + threadIdx.x;
    if (tid < 4096) {
        int which = tid >> 11;          // 0 = Q, 1 = K
        int rem   = tid & 2047;
        int r = rem >> 3, h = rem & 7;
        const float* W = which ? Wk : Wq;
        const float* p = W + r * DM + h * DK_;
        float s = 0.f;
        #pragma unroll
        for (int d = 0; d < DK_; ++d) s += p[d];
        ws[(which ? OFF_WK : OFF_WQ) + r * 8 + h] = s;
    } else if (tid < 4112) {
        int i = tid - 4096;             // 0..15
        int which = i >> 3, h = i & 7;
        const float* bb = which ? bk : bq;
        float s = 0.f;
        #pragma unroll
        for (int d = 0; d < DK_; ++d) s += bb[h * DK_ + d];
        ws[(which ? OFF_BK : OFF_BQ) + h] = s;
    }
}

// ---------------------------------------------------------------------------
// 2) corr_mean[b,l] = (1/8192) * sum_h (q_row.wq_sum[:,h]+bqs)*(k_row.wk_sum[:,h]+bks)
//    One thread per (b,l); reduced weights cached in LDS (16.06 KB).
// ---------------------------------------------------------------------------
__global__ void k_corr(const float* __restrict__ q, const float* __restrict__ key,
                       float* __restrict__ ws) {
    __shared__ float lds[4112];         // wq(2048) wk(2048) bq(8) bk(8)
    int t = threadIdx.x;
    for (int i = t; i < 4112; i += 256) lds[i] = ws[i];
    __syncthreads();

    int bl = blockIdx.x * 256 + t;      // 0..8191
    const float* qr = q   + (size_t)bl * DM;
    const float* kr = key + (size_t)bl * DM;
    float aq[8] = {0,0,0,0,0,0,0,0};
    float ak[8] = {0,0,0,0,0,0,0,0};
    for (int r = 0; r < DM; r += 4) {
        float4 q4 = *(const float4*)(qr + r);
        float4 k4 = *(const float4*)(kr + r);
        const float* wq = lds + r * 8;
        const float* wk = lds + 2048 + r * 8;
        #pragma unroll
        for (int h = 0; h < 8; ++h) {
            aq[h] += q4.x * wq[h] + q4.y * wq[8 + h] + q4.z * wq[16 + h] + q4.w * wq[24 + h];
            ak[h] += k4.x * wk[h] + k4.y * wk[8 + h] + k4.z * wk[16 + h] + k4.w * wk[24 + h];
        }
    }
    float s = 0.f;
    #pragma unroll
    for (int h = 0; h < 8; ++h)
        s += (aq[h] + lds[4096 + h]) * (ak[h] + lds[4104 + h]);
    ws[OFF_CORR + bl] = s * (1.0f / (float)(H_ * L_));
}

// ---------------------------------------------------------------------------
// 3) Per-batch top-6 selection + softmax. 8 lanes, trivial work.
// ---------------------------------------------------------------------------
__global__ void k_topk(float* __restrict__ ws) {
    int b = threadIdx.x;
    if (b >= B_) return;
    const float* corr = ws + OFF_CORR + b * L_;
    int*  idxs = (int*)(ws + OFF_IDX) + b * KTOP;
    float vals[KTOP]; int ids[KTOP];
    for (int kk = 0; kk < KTOP; ++kk) {
        float best = -INFINITY; int bi = 0;
        for (int l = 0; l < L_; ++l) {
            bool taken = false;
            for (int j = 0; j < kk; ++j) taken |= (ids[j] == l);
            float v = corr[l];
            if (!taken && v > best) { best = v; bi = l; }
        }
        vals[kk] = best; ids[kk] = bi;
    }
    float m = vals[0];                  // sorted desc -> vals[0] is max
    float e[KTOP], sum = 0.f;
    for (int kk = 0; kk < KTOP; ++kk) { e[kk] = __expf(vals[kk] - m); sum += e[kk]; }
    float inv = 1.0f / sum;
    for (int kk = 0; kk < KTOP; ++kk) {
        ws[OFF_W + b * KTOP + kk] = e[kk] * inv;
        idxs[kk] = ids[kk];
    }
}

// ---------------------------------------------------------------------------
// 4) agg[b,:] = sum_k w[b,k] * (values[b, idx_k, :] @ Wv + bv)
//    One block per batch, one thread per output column. Also zero pad rows 8..15.
// ---------------------------------------------------------------------------
__global__ void k_vagg(const float* __restrict__ values, const float* __restrict__ Wv,
                       const float* __restrict__ bv, float* __restrict__ ws) {
    int b = blockIdx.x, j = threadIdx.x;
    const int* idxs = (const int*)(ws + OFF_IDX) + b * KTOP;
    float acc = 0.f;
    for (int kk = 0; kk < KTOP; ++kk) {
        int idx  = idxs[kk];
        float w  = ws[OFF_W + b * KTOP + kk];
        const float* vrow = values + ((size_t)b * L_ + idx) * DM;   // broadcast reads
        float dot = 0.f;
        for (int r = 0; r < DM; ++r) dot += vrow[r] * Wv[r * DM + j];  // coalesced in j
        acc += w * (dot + bv[j]);
    }
    ws[OFF_AGG + b * DM + j] = acc;
    ws[OFF_AGG + (b + 8) * DM + j] = 0.0f;   // WMMA padding rows
}

// ---------------------------------------------------------------------------
// 5) out = agg(16x256, 8 real rows) @ Wp(256 x 262144) + bp  via V_WMMA_F32_16X16X4_F32.
//    One wave per 16-column tile; K looped in steps of 4. Wp streamed exactly once
//    (256 MB) -> HBM-bound, ~11 us floor at 23.3 TB/s.
//    A frag (16x4 f32): lane l -> M=l%16, VGPR j -> K=(l/16)*2+j   (from LDS)
//    B frag (4x16 f32): lane l -> N=l%16,  VGPR j -> K=(l/16)*2+j  (coalesced b32)
//    D frag: VGPR r, lanes 0-15 -> row r (batches 0..7); lanes 16-31 -> padding.
// ---------------------------------------------------------------------------
__global__ void k_out(const float* __restrict__ Wp, const float* __restrict__ bp,
                      const float* __restrict__ ws, float* __restrict__ out) {
    __shared__ float lagg[16 * DM];
    int t = threadIdx.x;
    for (int i = t; i < 16 * DM; i += 256) lagg[i] = ws[OFF_AGG + i];
    __syncthreads();

    int wave = t >> 5, lane = t & 31;
    int n0   = (blockIdx.x * 8 + wave) * 16;
    int ncol = lane & 15;
    int koff = (lane >> 4) * 2;
    int arow = ncol;                       // M = lane % 16

    float bpv = bp[n0 + ncol];
    v8f acc = {0.f, 0.f, 0.f, 0.f, 0.f, 0.f, 0.f, 0.f};
    const float* wpb = Wp + (size_t)n0 + ncol;

    for (int k0 = 0; k0 < DM; k0 += 4) {
        v2f a = *(const v2f*)(lagg + arow * DM + k0 + koff);   // ds_load_b64
        v2f b;
        const float* p = wpb + (size_t)(k0 + koff) * NOUT;
        b.x = p[0];
        b.y = p[NOUT];
        acc = __builtin_amdgcn_wmma_f32_16x16x4_f32(
            /*neg_a=*/false, a, /*neg_b=*/false, b,
            /*c_mod=*/(short)0, acc, /*reuse_a=*/false, /*reuse_b=*/false);
    }

    if (lane < 16) {                       // rows 0..7 live in lanes 0-15, VGPRs 0..7
        #pragma unroll
        for (int r = 0; r < 8; ++r)
            out[(size_t)r * NOUT + n0 + ncol] = acc[r] + bpv;
    }
}

// ---------------------------------------------------------------------------
extern "C" void kernel_launch(void* const* d_in, const int* in_sizes, int n_in,
                              void* d_out, int out_size, void* d_ws, size_t ws_size,
                              hipStream_t stream) {
    const float* queries = (const float*)d_in[0];
    const float* keys    = (const float*)d_in[1];
    const float* values  = (const float*)d_in[2];
    const float* Wq      = (const float*)d_in[3];
    const float* bq      = (const float*)d_in[4];
    const float* Wk      = (const float*)d_in[5];
    const float* bk      = (const float*)d_in[6];
    const float* Wv      = (const float*)d_in[7];
    const float* bv      = (const float*)d_in[8];
    const float* Wp      = (const float*)d_in[9];
    const float* bp      = (const float*)d_in[10];
    float* out = (float*)d_out;
    float* ws  = (float*)d_ws;

    k_colsum<<<17, 256, 0, stream>>>(Wq, bq, Wk, bk, ws);
    k_corr  <<<(B_ * L_) / 256, 256, 0, stream>>>(queries, keys, ws);
    k_topk  <<<1, 32, 0, stream>>>(ws);
    k_vagg  <<<B_, DM, 0, stream>>>(values, Wv, bv, ws);
    // 262144/16 = 16384 tiles, 8 waves (tiles) per 256-thread block -> 2048 blocks
    k_out   <<<NOUT / 16 / 8, 256, 0, stream>>>(Wp, bp, ws, out);
}